// SO2EquivariantGraphAttention_50208167690488
// MI455X (gfx1250) — compile-verified
//
#include <hip/hip_runtime.h>

// SO2 equivariant graph attention, CDNA5 (gfx1250), wave32, fp32 WMMA.
//
// Key observation: the equivariant output of the FIRST so2_conv is dead in the
// reference (msg is rebuilt from gating parts before use). Only x0_extra
// (first 576 cols of x0 @ c1_wm0) is live -> c1_wm1/c1_wm2 and the m=1/m=2
// paths of conv1 are skipped, wigner is applied only to rows {0,2,6}, and the
// radial MLP only needs its first 384 output columns.
//
// All shared-weight batched GEMMs run on V_WMMA_F32_16X16X4_F32 (fp32 tensor
// path, exact fp32 accumulation). Weights (<4MB) and x (23MB) are L2-resident
// on the 192MB L2, so the GEMMs are matrix-pipe bound, not HBM bound.
//
// Workspace layout (floats), total ~152 MB (assumes ws_size is sufficient):
//   attnOut  : NE*320
//   alphaBuf : NE*8        (logits, then exp() after K2)
//   segMaxU  : NN*8 (uint, order-encoded float max)
//   denom    : NN*8
//   node     : NN*9*128    (atomic accumulation)

#define NE 80000
#define NN 10000

typedef __attribute__((ext_vector_type(2))) float v2f;
typedef __attribute__((ext_vector_type(8))) float v8f;

// ---- order-preserving float <-> uint encoding for atomicMax on floats ----
__device__ __forceinline__ unsigned f2ord(float f) {
  unsigned u = __float_as_uint(f);
  return (u >> 31) ? ~u : (u | 0x80000000u);
}
__device__ __forceinline__ float ord2f(unsigned u) {
  return (u >> 31) ? __uint_as_float(u & 0x7FFFFFFFu) : __uint_as_float(~u);
}

__device__ __forceinline__ float siluf(float x) { return x / (1.0f + expf(-x)); }

// ---- fp32 WMMA 16x16 tile GEMM: C(16x16) += A(16xK) * B(KxN tile) ----------
// A supplied via functor afn(m,k) reading LDS; B row-major in global memory.
// Fragment layouts per CDNA5 ISA 7.12.2:
//   A 16x4 : lane&15 = M, half=lane>>4 selects K pair {2h, 2h+1}
//   B 4x16 : lane&15 = N, half selects K pair
//   C      : VGPR v holds row m = v + 8*half, col n = nb + (lane&15)
template <typename AFn>
__device__ __forceinline__ v8f wmma_gemm_tile(AFn afn, const float* __restrict__ B,
                                              int ldB, int nb, int Kdim) {
  const int col  = threadIdx.x & 15;
  const int half = (threadIdx.x >> 4) & 1;
  v8f acc = {};
#pragma unroll 4
  for (int k = 0; k < Kdim; k += 4) {
    const int ka = k + 2 * half;
    v2f a, b;
    a.x = afn(col, ka);
    a.y = afn(col, ka + 1);
    b.x = B[(size_t)ka * ldB + nb + col];
    b.y = B[(size_t)(ka + 1) * ldB + nb + col];
    acc = __builtin_amdgcn_wmma_f32_16x16x4_f32(false, a, false, b, (short)0, acc,
                                                false, false);
  }
  return acc;
}

// Compile-time bias / row-clamp specialization: the bias element bias[nb+col]
// is invariant across the 8 accumulator rows -> load it exactly once; the row
// clamp folds to a constant predicate per half-wave.
template <int MMAX, bool HAS_BIAS>
__device__ __forceinline__ void wmma_store_tile(float* dst, int ld, int nb, v8f acc,
                                                const float* __restrict__ bias) {
  const int col  = threadIdx.x & 15;
  const int half = (threadIdx.x >> 4) & 1;
  float bv = 0.0f;
  if (HAS_BIAS) bv = bias[nb + col];
#pragma unroll
  for (int v = 0; v < 8; ++v) {
    const int m = v + 8 * half;
    if (m < MMAX) dst[m * ld + nb + col] = acc[v] + bv;
  }
}

// ============================================================================
// K1: per-edge phase 1 (16 edges / block, 8 waves).
//  x_edge -> GEMM(320->128) -> LN+scaled_silu -> GEMM(128->384) = rad
//  wigner rows {0,2,6} (fused * rad) -> GEMM(384->576) = x0_extra
//  -> per-head LN + smooth_leaky_relu + alpha logits + atomicMax(seg max)
//  -> spill attn_out
// ============================================================================
__global__ __launch_bounds__(256) void so2ega_k1(
    const float* __restrict__ x, const int* __restrict__ an,
    const float* __restrict__ edist, const int* __restrict__ eidx,
    const float* __restrict__ wigner, const float* __restrict__ src_emb,
    const float* __restrict__ tgt_emb, const float* __restrict__ rad_w1,
    const float* __restrict__ rad_b1, const float* __restrict__ rad_lnw,
    const float* __restrict__ rad_lnb, const float* __restrict__ rad_w2,
    const float* __restrict__ rad_b2, const float* __restrict__ c1_wm0,
    const float* __restrict__ c1_bm0, const float* __restrict__ ln_a_w,
    const float* __restrict__ ln_a_b, const float* __restrict__ alpha_dot,
    float* __restrict__ attnOut, float* __restrict__ alphaBuf,
    unsigned* __restrict__ segMaxU) {
  __shared__ float P[15360];      // pooled LDS, stages carved below
  __shared__ float sWig[16 * 27]; // wigner rows {0,2,6}
  __shared__ int sSrc[16], sTgt[16];

  const int tid  = threadIdx.x;
  const int wave = tid >> 5;
  const int e0   = blockIdx.x * 16;

  float* xe = P;          // 16x320  (stage A)
  float* t1 = P + 5120;   // 16x128  (stage A/B)
  float* rr = P + 7168;   // 16x384  (stage B/C)
  float* mw = P;          // 16x384  (stage C, xe dead)
  float* y0 = P + 6144;   // 16x576  (stage D, rr dead)

  if (tid < 16) {
    sSrc[tid] = eidx[e0 + tid];
    sTgt[tid] = eidx[NE + e0 + tid];
  }
  __syncthreads();

  // ---- gather x_edge = [edge_distance | src_emb[an[src]] | tgt_emb[an[tgt]]]
  for (int i = tid; i < 16 * 320; i += 256) {
    const int e = i / 320, c = i % 320;
    float v;
    if (c < 64)       v = edist[(size_t)(e0 + e) * 64 + c];
    else if (c < 192) v = src_emb[an[sSrc[e]] * 128 + (c - 64)];
    else              v = tgt_emb[an[sTgt[e]] * 128 + (c - 192)];
    xe[e * 320 + c] = v;
  }
  // wigner rows {0,2,6}: rowsel(j3) = (2<<j3)-2
  for (int i = tid; i < 16 * 27; i += 256) {
    const int e = i / 27, rem = i % 27, j3 = rem / 9, j = rem % 9;
    sWig[i] = wigner[(size_t)(e0 + e) * 81 + ((2 << j3) - 2) * 9 + j];
  }
  __syncthreads();

  // ---- GEMM1: t1 = xe @ rad_w1 + rad_b1  (M16, K320, N128) ----
  for (int t = wave; t < 8; t += 8) {
    const int nb = t * 16;
    auto afn = [&](int m, int k) { return xe[m * 320 + k]; };
    v8f acc  = wmma_gemm_tile(afn, rad_w1, 128, nb, 320);
    wmma_store_tile<16, true>(t1, 128, nb, acc, rad_b1);
  }
  __syncthreads();

  // ---- LN(128) + scaled_silu, 2 rows per wave ----
  {
    const int lane = tid & 31;
    for (int r = wave; r < 16; r += 8) {
      float vals[4], s = 0.f, sq = 0.f;
#pragma unroll
      for (int q = 0; q < 4; ++q) {
        float v = t1[r * 128 + lane + q * 32];
        vals[q] = v; s += v; sq += v * v;
      }
      for (int o = 16; o >= 1; o >>= 1) {
        s  += __shfl_xor(s, o, 32);
        sq += __shfl_xor(sq, o, 32);
      }
      const float mean = s * (1.0f / 128.0f);
      const float var  = sq * (1.0f / 128.0f) - mean * mean;
      const float inv  = rsqrtf(var + 1e-5f);
#pragma unroll
      for (int q = 0; q < 4; ++q) {
        const int c = lane + q * 32;
        float v = (vals[q] - mean) * inv * rad_lnw[c] + rad_lnb[c];
        t1[r * 128 + c] = siluf(v) * (1.0f / 0.6f);
      }
    }
  }
  __syncthreads();

  // ---- GEMM2: rr = t1 @ rad_w2[:, :384] + rad_b2  (M16, K128, N384 of 768) --
  for (int t = wave; t < 24; t += 8) {
    const int nb = t * 16;
    auto afn = [&](int m, int k) { return t1[m * 128 + k]; };
    v8f acc  = wmma_gemm_tile(afn, rad_w2, 768, nb, 128);
    wmma_store_tile<16, true>(rr, 384, nb, acc, rad_b2);
  }
  __syncthreads();

  // ---- wigner rows {0,2,6} on concat(x_src, x_tgt), fused radial scaling ----
  for (int i = tid; i < 16 * 384; i += 256) {
    const int e = i / 384, rem = i % 384, j3 = rem / 128, c = rem & 127;
    const float* wg = &sWig[e * 27 + j3 * 9];
    const float* xs = (c < 64) ? (x + (size_t)sSrc[e] * 576 + c)
                               : (x + (size_t)sTgt[e] * 576 + (c - 64));
    float acc = 0.f;
#pragma unroll
    for (int j = 0; j < 9; ++j) acc += wg[j] * xs[j * 64];
    mw[i] = acc * rr[e * 384 + rem];
  }
  __syncthreads();

  // ---- GEMM3: y0 = mw @ c1_wm0[:, :576] + c1_bm0  (M16, K384, N576 of 768) --
  for (int t = wave; t < 36; t += 8) {
    const int nb = t * 16;
    auto afn = [&](int m, int k) { return mw[m * 384 + k]; };
    v8f acc  = wmma_gemm_tile(afn, c1_wm0, 768, nb, 384);
    wmma_store_tile<16, true>(y0, 576, nb, acc, c1_bm0);
  }
  __syncthreads();

  // ---- attention extras: a = y0[:, :256], attn_out = y0[:, 256:576] ----
  if (tid < 128) {
    const int e = tid >> 3, h = tid & 7;
    const float* a = &y0[e * 576 + h * 32];
    float s = 0.f, sq = 0.f;
#pragma unroll
    for (int k = 0; k < 32; ++k) { float v = a[k]; s += v; sq += v * v; }
    const float mean = s * (1.0f / 32.0f);
    const float var  = sq * (1.0f / 32.0f) - mean * mean;
    const float inv  = rsqrtf(var + 1e-5f);
    float dot = 0.f;
#pragma unroll
    for (int k = 0; k < 32; ++k) {
      float v   = (a[k] - mean) * inv * ln_a_w[k] + ln_a_b[k];
      float sig = 1.0f / (1.0f + expf(-v));
      float sl  = 0.6f * v + 0.4f * v * (2.0f * sig - 1.0f); // smooth_leaky(0.2)
      dot += sl * alpha_dot[h * 32 + k];
    }
    alphaBuf[(size_t)(e0 + e) * 8 + h] = dot;
    atomicMax(&segMaxU[sTgt[e] * 8 + h], f2ord(dot));
  }
  for (int i = tid; i < 16 * 320; i += 256) {
    const int e = i / 320, c = i % 320;
    attnOut[(size_t)(e0 + e) * 320 + c] = y0[e * 576 + 256 + c];
  }
}

// ============================================================================
// K2: softmax pass 2 : ex = exp(alpha - segmax[tgt]); denom += ex
// ============================================================================
__global__ __launch_bounds__(256) void so2ega_k2(
    const int* __restrict__ eidx, const unsigned* __restrict__ segMaxU,
    float* __restrict__ alphaBuf, float* __restrict__ denom) {
  const int i = blockIdx.x * 256 + threadIdx.x;
  if (i >= NE * 8) return;
  const int e = i >> 3, h = i & 7;
  const int t = eidx[NE + e];
  const float m  = ord2f(segMaxU[t * 8 + h]);
  const float ex = expf(alphaBuf[i] - m);
  alphaBuf[i] = ex;
  atomicAdd(&denom[t * 8 + h], ex);
}

// ============================================================================
// K3: per-edge phase 2 (4 edges / block).
//  alpha normalize -> gating (h, tp VALU; Xp via WMMA M=32) -> msg2
//  -> so2_conv2 (3 WMMA GEMMs + complex combine) -> alpha scale
//  -> wigner_inv -> atomicAdd into node accumulator
// ============================================================================
__global__ __launch_bounds__(256) void so2ega_k3(
    const float* __restrict__ x, const int* __restrict__ eidx,
    const float* __restrict__ t_ij, const float* __restrict__ rl_ij,
    const float* __restrict__ wigner_inv, const float* __restrict__ gata_wh,
    const float* __restrict__ gata_wx, const float* __restrict__ gata_wt,
    const float* __restrict__ c2_wm0, const float* __restrict__ c2_bm0,
    const float* __restrict__ c2_wm1, const float* __restrict__ c2_wm2,
    const float* __restrict__ attnOut, const float* __restrict__ alphaBuf,
    const float* __restrict__ denom, float* __restrict__ node) {
  __shared__ float msg2[4 * 576];   // gated input, 9x64 per edge
  __shared__ float msg3[4 * 1152];  // conv2 output, 9x128 per edge
  __shared__ float scr[4096];       // X_j+Xp, then y0/y1/y2 GEMM outputs
  __shared__ float sH[4 * 64], sTp[4 * 64], sAl[4 * 8], sAm[4], sRl[4 * 8];
  __shared__ float sWinv[4 * 81];
  __shared__ int sTgt[4];

  const int tid  = threadIdx.x;
  const int wave = tid >> 5;
  const int e0   = blockIdx.x * 4;

  if (tid < 4) sTgt[tid] = eidx[NE + e0 + tid];
  __syncthreads();

  if (tid < 32) {
    const int e = tid >> 3, h = tid & 7;
    const float ex = alphaBuf[(size_t)(e0 + e) * 8 + h];
    sAl[tid] = ex / (denom[sTgt[e] * 8 + h] + 1e-16f);
    sRl[tid] = rl_ij[(size_t)(e0 + e) * 8 + h];
  }
  for (int i = tid; i < 4 * 81; i += 256)
    sWinv[i] = wigner_inv[(size_t)(e0 + i / 81) * 81 + i % 81];
  // X_j = x[tgt, 1:, :] -> 512 contiguous floats per edge
  for (int i = tid; i < 4 * 512; i += 256)
    scr[i] = x[(size_t)sTgt[i >> 9] * 576 + 64 + (i & 511)];
  __syncthreads();

  if (tid < 4) {
    float s = 0.f;
#pragma unroll
    for (int h = 0; h < 8; ++h) s += sAl[tid * 8 + h];
    sAm[tid] = s * 0.125f;
  }
  // h = silu(h_j @ gata_wh), tp = silu(t_ij @ gata_wt)  (256 outputs each, VALU)
  {
    const int e = tid >> 6, o = tid & 63;
    const float* xh = x + (size_t)sTgt[e] * 576;
    float a = 0.f;
    for (int c = 0; c < 64; ++c) a += xh[c] * gata_wh[c * 64 + o];
    sH[tid] = siluf(a);
    const float* tj = t_ij + (size_t)(e0 + e) * 128;
    float b = 0.f;
    for (int c = 0; c < 128; ++c) b += tj[c] * gata_wt[c * 64 + o];
    sTp[tid] = siluf(b);
  }
  __syncthreads();

  // ---- Xp = X_j @ gata_wx  (M32 = 2 tiles, K64, N64) ----
  for (int t = wave; t < 8; t += 8) {
    const int mt = t >> 2, nb = (t & 3) * 16;
    auto afn = [&](int m, int k) { return scr[(mt * 16 + m) * 64 + k]; };
    v8f acc  = wmma_gemm_tile(afn, gata_wx, 64, nb, 64);
    wmma_store_tile<16, false>(scr + 2048 + mt * 16 * 64, 64, nb, acc, nullptr);
  }
  __syncthreads();

  // ---- build msg2 (9 rows x 64 per edge) ----
  for (int i = tid; i < 4 * 576; i += 256) {
    const int e = i / 576, rem = i % 576, row = rem >> 6, c = rem & 63;
    const float* at = attnOut + (size_t)(e0 + e) * 320;
    const float am  = sAm[e];
    float v;
    if (row == 0) {
      v = am * at[c] * sH[e * 64 + c];
    } else {
      const int l    = (row <= 3) ? 0 : 1;
      const float od = am * at[64 + l * 64 + c];
      const float ot = am * at[192 + l * 64 + c];
      const float xp = scr[2048 + (e * 8 + row - 1) * 64 + c];
      v = od * xp + ot * sTp[e * 64 + c] * sRl[e * 8 + row - 1];
    }
    msg2[i] = v;
  }
  __syncthreads();

  // ---- conv2 x0: rows {0,2,6} -> (M4, K192, N384) + bias ----
  for (int t = wave; t < 24; t += 8) {
    const int nb = t * 16;
    auto afn = [&](int m, int k) {
      return msg2[(m & 3) * 576 + ((2 << (k >> 6)) - 2) * 64 + (k & 63)];
    };
    v8f acc = wmma_gemm_tile(afn, c2_wm0, 384, nb, 192);
    wmma_store_tile<4, true>(scr, 384, nb, acc, c2_bm0);
  }
  __syncthreads();
  for (int i = tid; i < 4 * 384; i += 256) { // out0 -> final rows {0,2,6}
    const int e = i / 384, rem = i % 384, j = rem >> 7, c = rem & 127;
    msg3[e * 1152 + ((2 << j) - 2) * 128 + c] = scr[e * 384 + j * 128 + c];
  }
  __syncthreads();

  // ---- conv2 x1: comp0=[o3|o7], comp1=[o1|o5] -> (M8, K128, N512) ----
  for (int t = wave; t < 32; t += 8) {
    const int nb = t * 16;
    auto afn = [&](int m, int k) {
      const int e = (m >> 1) & 3, comp = m & 1;
      const int og = (k < 64) ? (comp ? 1 : 3) : (comp ? 5 : 7);
      return msg2[e * 576 + og * 64 + (k & 63)];
    };
    v8f acc = wmma_gemm_tile(afn, c2_wm1, 512, nb, 128);
    wmma_store_tile<8, false>(scr, 512, nb, acc, nullptr);
  }
  __syncthreads();
  for (int i = tid; i < 4 * 256; i += 256) { // complex combine -> rows {3,7,1,5}
    const int e = i >> 8, cc = i & 255, hh = cc >> 7, c = cc & 127;
    const float outr = scr[(2 * e) * 512 + cc] - scr[(2 * e + 1) * 512 + 256 + cc];
    const float outi = scr[(2 * e + 1) * 512 + cc] + scr[(2 * e) * 512 + 256 + cc];
    msg3[e * 1152 + (hh ? 7 : 3) * 128 + c] = outr;
    msg3[e * 1152 + (hh ? 5 : 1) * 128 + c] = outi;
  }
  __syncthreads();

  // ---- conv2 x2: comp0=o8, comp1=o4 -> (M8, K64, N256) ----
  for (int t = wave; t < 16; t += 8) {
    const int nb = t * 16;
    auto afn = [&](int m, int k) {
      const int e = (m >> 1) & 3, comp = m & 1;
      return msg2[e * 576 + (comp ? 4 : 8) * 64 + k];
    };
    v8f acc = wmma_gemm_tile(afn, c2_wm2, 256, nb, 64);
    wmma_store_tile<8, false>(scr, 256, nb, acc, nullptr);
  }
  __syncthreads();
  for (int i = tid; i < 4 * 128; i += 256) { // combine -> rows {8,4}
    const int e = i >> 7, c = i & 127;
    msg3[e * 1152 + 8 * 128 + c] =
        scr[(2 * e) * 256 + c] - scr[(2 * e + 1) * 256 + 128 + c];
    msg3[e * 1152 + 4 * 128 + c] =
        scr[(2 * e + 1) * 256 + c] + scr[(2 * e) * 256 + 128 + c];
  }
  __syncthreads();

  // ---- per-head attention scaling ----
  for (int i = tid; i < 4 * 1152; i += 256)
    msg3[i] *= sAl[(i / 1152) * 8 + ((i & 127) >> 4)];
  __syncthreads();

  // ---- wigner_inv rotate + scatter-add into node accumulator ----
  for (int i = tid; i < 4 * 1152; i += 256) {
    const int e = i / 1152, rem = i % 1152, r = rem >> 7, c = rem & 127;
    const float* wg = &sWinv[e * 81 + r * 9];
    const float* m3 = &msg3[e * 1152 + c];
    float acc = 0.f;
#pragma unroll
    for (int j = 0; j < 9; ++j) acc += wg[j] * m3[j * 128];
    atomicAdd(&node[(size_t)sTgt[e] * 1152 + rem], acc);
  }
}

// ============================================================================
// K4: per-node projection  out[n,k,o] = sum_i node[n,k,i]*proj_w[L_IDX[k],o,i]
// ============================================================================
__global__ __launch_bounds__(256) void so2ega_k4(
    const float* __restrict__ node, const float* __restrict__ proj_w,
    const float* __restrict__ proj_b, float* __restrict__ out) {
  __shared__ float sN[1152];
  const int n = blockIdx.x, tid = threadIdx.x;
  for (int i = tid; i < 1152; i += 256) sN[i] = node[(size_t)n * 1152 + i];
  __syncthreads();
  for (int t = tid; t < 576; t += 256) {
    const int k = t >> 6, o = t & 63;
    const int l = (k == 0) ? 0 : ((k < 4) ? 1 : 2); // L_IDX
    const float* w = proj_w + l * 8192 + o * 128;
    const float* v = sN + k * 128;
    float acc = (k == 0) ? proj_b[o] : 0.f;
#pragma unroll 4
    for (int i = 0; i < 128; ++i) acc += v[i] * w[i];
    out[(size_t)n * 576 + t] = acc;
  }
}

// ============================================================================
extern "C" void kernel_launch(void* const* d_in, const int* in_sizes, int n_in,
                              void* d_out, int out_size, void* d_ws,
                              size_t ws_size, hipStream_t stream) {
  const float* x         = (const float*)d_in[0];
  const int*   an        = (const int*)d_in[1];
  const float* edist     = (const float*)d_in[2];
  const int*   eidx      = (const int*)d_in[3];
  const float* t_ij      = (const float*)d_in[4];
  const float* rl_ij     = (const float*)d_in[5];
  const float* wigner    = (const float*)d_in[6];
  const float* wigner_iv = (const float*)d_in[7];
  const float* src_emb   = (const float*)d_in[8];
  const float* tgt_emb   = (const float*)d_in[9];
  const float* rad_w1    = (const float*)d_in[10];
  const float* rad_b1    = (const float*)d_in[11];
  const float* rad_lnw   = (const float*)d_in[12];
  const float* rad_lnb   = (const float*)d_in[13];
  const float* rad_w2    = (const float*)d_in[14];
  const float* rad_b2    = (const float*)d_in[15];
  const float* c1_wm0    = (const float*)d_in[16];
  const float* c1_bm0    = (const float*)d_in[17];
  // d_in[18], d_in[19] (c1_wm1, c1_wm2): dead in reference graph
  const float* ln_a_w    = (const float*)d_in[20];
  const float* ln_a_b    = (const float*)d_in[21];
  const float* alpha_dot = (const float*)d_in[22];
  const float* gata_wh   = (const float*)d_in[23];
  const float* gata_wx   = (const float*)d_in[24];
  const float* gata_wt   = (const float*)d_in[25];
  const float* c2_wm0    = (const float*)d_in[26];
  const float* c2_bm0    = (const float*)d_in[27];
  const float* c2_wm1    = (const float*)d_in[28];
  const float* c2_wm2    = (const float*)d_in[29];
  const float* proj_w    = (const float*)d_in[30];
  const float* proj_b    = (const float*)d_in[31];

  float* ws        = (float*)d_ws;
  float* attnOut   = ws;                                     // NE*320
  float* alphaBuf  = attnOut + (size_t)NE * 320;             // NE*8
  unsigned* segU   = (unsigned*)(alphaBuf + (size_t)NE * 8); // NN*8
  float* denomB    = (float*)(segU + NN * 8);                // NN*8
  float* nodeAcc   = denomB + NN * 8;                        // NN*1152

  hipMemsetAsync(segU, 0, (size_t)NN * 8 * sizeof(unsigned), stream);
  hipMemsetAsync(denomB, 0, (size_t)NN * 8 * sizeof(float), stream);
  hipMemsetAsync(nodeAcc, 0, (size_t)NN * 1152 * sizeof(float), stream);

  so2ega_k1<<<NE / 16, 256, 0, stream>>>(
      x, an, edist, eidx, wigner, src_emb, tgt_emb, rad_w1, rad_b1, rad_lnw,
      rad_lnb, rad_w2, rad_b2, c1_wm0, c1_bm0, ln_a_w, ln_a_b, alpha_dot,
      attnOut, alphaBuf, segU);

  so2ega_k2<<<(NE * 8 + 255) / 256, 256, 0, stream>>>(eidx, segU, alphaBuf,
                                                      denomB);

  so2ega_k3<<<NE / 4, 256, 0, stream>>>(
      x, eidx, t_ij, rl_ij, wigner_iv, gata_wh, gata_wx, gata_wt, c2_wm0,
      c2_bm0, c2_wm1, c2_wm2, attnOut, alphaBuf, denomB, nodeAcc);

  so2ega_k4<<<NN, 256, 0, stream>>>(nodeAcc, proj_w, proj_b, (float*)d_out);
}